// BaseRCIEncoder_41893111005431
// MI455X (gfx1250) — compile-verified
//
#include <hip/hip_runtime.h>
#include <math.h>

// Problem constants (match reference)
#define Hd  64
#define NHd 4
#define Ld  2
#define Nn  20000
#define Ee  320000
#define Bb  64
#define SLOPE 0.2f

typedef __attribute__((ext_vector_type(2))) float v2f;
typedef __attribute__((ext_vector_type(8))) float v8f;

__device__ __forceinline__ v8f wmma4(v2f a, v2f b, v8f c) {
  // D = A(16x4,f32) x B(4x16,f32) + C(16x16,f32)
  return __builtin_amdgcn_wmma_f32_16x16x4_f32(
      /*neg_a=*/false, a, /*neg_b=*/false, b,
      /*c_mod=*/(short)0, c, /*reuse_a=*/false, /*reuse_b=*/false);
}

__device__ __forceinline__ float elu1(float f)   { return f > 0.f ? f : expm1f(f); }
__device__ __forceinline__ float lrelu(float f)  { return f > 0.f ? f : SLOPE * f; }

// Ordered-uint encoding so float max == uint max (handles negatives)
__device__ __forceinline__ unsigned fenc(float f) {
  unsigned u = __float_as_uint(f);
  return (u & 0x80000000u) ? ~u : (u | 0x80000000u);
}
__device__ __forceinline__ float fdec(unsigned u) {
  unsigned b = (u & 0x80000000u) ? (u ^ 0x80000000u) : ~u;
  return __uint_as_float(b);
}

// ---------------------------------------------------------------- utilities
__global__ void zero_u32_kernel(unsigned* __restrict__ p, int n) {
  int i = blockIdx.x * blockDim.x + threadIdx.x;
  int stride = gridDim.x * blockDim.x;
  for (; i < n; i += stride) p[i] = 0u;
}

// ------------------------------------------------- weight fragment packing
// Pack W[K][NC] (row-major) into per-lane B-fragment order for
// V_WMMA_F32_16X16X4_F32: one float4 = fragments for k-steps (8kk, 8kk+4).
__global__ void pack_w_kernel(const float* __restrict__ W, float* __restrict__ Wp,
                              int K, int NC) {
  int i = blockIdx.x * blockDim.x + threadIdx.x;  // one float4 each
  int total = (NC / 16) * (K / 8) * 32;
  if (i >= total) return;
  int lane = i & 31;
  int kk   = (i >> 5) % (K / 8);
  int t    = (i >> 5) / (K / 8);
  int mrow = lane & 15, kh = lane >> 4;
  int col = t * 16 + mrow;
  int k = kk * 8;
  float4 v;
  v.x = W[(k + 2 * kh)     * NC + col];
  v.y = W[(k + 2 * kh + 1) * NC + col];
  v.z = W[(k + 4 + 2 * kh)     * NC + col];
  v.w = W[(k + 4 + 2 * kh + 1) * NC + col];
  ((float4*)Wp)[i] = v;
}

// Same, for the virtual row-concatenation [Wni; Wnj; Wf]  (K=192, NC=256)
__global__ void pack_wcat_kernel(const float* __restrict__ Wni, const float* __restrict__ Wnj,
                                 const float* __restrict__ Wf, float* __restrict__ Wp) {
  const int K = 192, NC = 256;
  int i = blockIdx.x * blockDim.x + threadIdx.x;
  int total = (NC / 16) * (K / 8) * 32;
  if (i >= total) return;
  int lane = i & 31;
  int kk   = (i >> 5) % (K / 8);
  int t    = (i >> 5) / (K / 8);
  int mrow = lane & 15, kh = lane >> 4;
  int col = t * 16 + mrow;
  int k = kk * 8;
  int r[4] = { k + 2 * kh, k + 2 * kh + 1, k + 4 + 2 * kh, k + 5 + 2 * kh };
  float v[4];
  for (int j = 0; j < 4; ++j) {
    int rr = r[j];
    v[j] = (rr < 64) ? Wni[rr * NC + col]
         : (rr < 128) ? Wnj[(rr - 64) * NC + col]
                      : Wf[(rr - 128) * NC + col];
  }
  float4 q; q.x = v[0]; q.y = v[1]; q.z = v[2]; q.w = v[3];
  ((float4*)Wp)[i] = q;
}

// ------------------------------------------------------------ input encoders
__global__ void fp_enc_kernel(const float* __restrict__ fp, const float* __restrict__ W,
                              const float* __restrict__ b, float* __restrict__ out) {
  int i = blockIdx.x * blockDim.x + threadIdx.x;
  if (i >= Bb * Hd) return;
  int g = i >> 6, h = i & 63;
  const float* fr = fp + (size_t)g * 2048;
  float s = b[h];
  for (int k = 0; k < 2048; ++k) s += fr[k] * W[k * Hd + h];
  out[i] = elu1(s);
}

__global__ void node_init_kernel(const float* __restrict__ x, const int* __restrict__ rc,
                                 const float* __restrict__ W, const float* __restrict__ b,
                                 const float* __restrict__ emb, float* __restrict__ nf) {
  int i = blockIdx.x * blockDim.x + threadIdx.x;
  if (i >= Nn * Hd) return;
  int n = i >> 6, h = i & 63;
  const float* xr = x + (size_t)n * 81;
  float s = b[h] + emb[rc[n] * Hd + h];
  for (int k = 0; k < 81; ++k) s += xr[k] * W[k * Hd + h];
  nf[i] = s;
}

__global__ void edge_init_kernel(const float* __restrict__ e, const float* __restrict__ W,
                                 const float* __restrict__ b, float* __restrict__ ef) {
  int i = blockIdx.x * blockDim.x + threadIdx.x;
  if (i >= Ee * Hd) return;
  int ei = i >> 6, h = i & 63;
  const float* er = e + (size_t)ei * 17;
  float s = b[h];
  for (int k = 0; k < 17; ++k) s += er[k] * W[k * Hd + h];
  ef[i] = s;
}

// --------------------------------------------------------- fused edge kernel
// Per block (128 thr / 4 waves): 16 edges.
//  A = [nf[src] | nf[dst] | ef]  (16x192, LDS)
//  f_out(16x256) = leaky_relu(A @ WcatP) kept in LDS  (WMMA f32 16x16x4)
//  logits(16x4)  = <f_out, attn> -> global + atomicMax segment-max
//  new_e(16x64)  = elu(f_out @ WmeP + bme) + ef   (second WMMA pass, K=256)
__global__ __launch_bounds__(128)
void egat_edge_kernel(const float* __restrict__ nf, const float* __restrict__ ef,
                      const int* __restrict__ src, const int* __restrict__ dst,
                      const float4* __restrict__ WcatP, const float* __restrict__ attn,
                      const float4* __restrict__ WmeP, const float* __restrict__ bme,
                      float* __restrict__ ef_out, float* __restrict__ logits,
                      unsigned* __restrict__ mmax) {
  __shared__ float As[16 * 192];
  __shared__ float Fs[16 * 256];
  const int tid = threadIdx.x;
  const int e0  = blockIdx.x * 16;

  // Cooperative gather: coalesced 256B row segments
  for (int idx = tid; idx < 16 * 192; idx += 128) {
    int row = idx / 192, col = idx - row * 192;
    int e = e0 + row;
    float v;
    if (col < 64)       v = nf[(size_t)src[e] * Hd + col];
    else if (col < 128) v = nf[(size_t)dst[e] * Hd + (col - 64)];
    else                v = ef[(size_t)e * Hd + (col - 128)];
    As[idx] = v;
  }
  __syncthreads();

  const int lane = tid & 31;
  const int wave = tid >> 5;
  const int mrow = lane & 15;   // A: M row / B,C: N column offset
  const int kh   = lane >> 4;   // K-pair / M-half selector

  // f_out: wave w owns columns [64w, 64w+64)  (== head w)
  for (int t = 0; t < 4; ++t) {
    const int tg  = wave * 4 + t;          // global 16-col tile index
    const int col = tg * 16 + mrow;
    const float4* wp = WcatP + ((size_t)tg * 24) * 32 + lane;  // K=192 -> 24 float4 steps
    v8f acc = {};
    const float* Ap = &As[mrow * 192 + 2 * kh];
    for (int kk = 0; kk < 24; ++kk) {
      float4 bq = wp[(size_t)kk * 32];
      v2f a0 = *(const v2f*)(Ap + 8 * kk);
      v2f a1 = *(const v2f*)(Ap + 8 * kk + 4);
      v2f b0; b0.x = bq.x; b0.y = bq.y;
      v2f b1; b1.x = bq.z; b1.y = bq.w;
      acc = wmma4(a0, b0, acc);
      acc = wmma4(a1, b1, acc);
    }
    for (int r = 0; r < 8; ++r)
      Fs[(r + 8 * kh) * 256 + col] = lrelu(acc[r]);
  }
  __syncthreads();

  // logits + segment-max (threads 0..63 -> 16 edges x 4 heads)
  if (tid < 64) {
    int er = tid >> 2, hh = tid & 3;
    const float* av = attn + hh * Hd;
    const float* fr = &Fs[er * 256 + hh * Hd];
    float s = 0.f;
    for (int j = 0; j < Hd; ++j) s += fr[j] * av[j];
    int e = e0 + er;
    logits[(size_t)e * NHd + hh] = s;
    atomicMax(&mmax[(size_t)dst[e] * NHd + hh], fenc(s));
  }

  // new_e: wave w owns output cols [16w,16w+16), K = 256 from Fs
  {
    const int col = wave * 16 + mrow;
    const float4* wp = WmeP + ((size_t)wave * 32) * 32 + lane;  // K=256 -> 32 steps
    v8f acc = {};
    const float* Ap = &Fs[mrow * 256 + 2 * kh];
    for (int kk = 0; kk < 32; ++kk) {
      float4 bq = wp[(size_t)kk * 32];
      v2f a0 = *(const v2f*)(Ap + 8 * kk);
      v2f a1 = *(const v2f*)(Ap + 8 * kk + 4);
      v2f b0; b0.x = bq.x; b0.y = bq.y;
      v2f b1; b1.x = bq.z; b1.y = bq.w;
      acc = wmma4(a0, b0, acc);
      acc = wmma4(a1, b1, acc);
    }
    for (int r = 0; r < 8; ++r) {
      int e = e0 + r + 8 * kh;
      float f = elu1(acc[r] + bme[col]);
      ef_out[(size_t)e * Hd + col] = f + ef[(size_t)e * Hd + col];
    }
  }
}

// exp(logit - m[dst]) and segment-sum denominator
__global__ void softmax_exp_kernel(const int* __restrict__ dst, const unsigned* __restrict__ mmax,
                                   float* __restrict__ exb, float* __restrict__ denom) {
  int i = blockIdx.x * blockDim.x + threadIdx.x;
  if (i >= Ee * NHd) return;
  int e = i >> 2, hh = i & 3;
  int d = dst[e];
  float m = fdec(mmax[(size_t)d * NHd + hh]);
  float ex = __expf(exb[i] - m);
  exb[i] = ex;
  atomicAdd(&denom[(size_t)d * NHd + hh], ex);
}

// h = nf @ fcnW  (M=N nodes, K=64, N=256)
__global__ __launch_bounds__(128)
void node_h_kernel(const float* __restrict__ nf, const float4* __restrict__ Wp,
                   float* __restrict__ h) {
  __shared__ float As[16 * 64];
  const int tid = threadIdx.x;
  const int r0 = blockIdx.x * 16;
  for (int idx = tid; idx < 16 * 64; idx += 128) As[idx] = nf[(size_t)r0 * Hd + idx];
  __syncthreads();
  const int lane = tid & 31, wave = tid >> 5, mrow = lane & 15, kh = lane >> 4;
  for (int t = 0; t < 4; ++t) {
    const int tg  = wave * 4 + t;
    const int col = tg * 16 + mrow;
    const float4* wp = Wp + ((size_t)tg * 8) * 32 + lane;      // K=64 -> 8 steps
    v8f acc = {};
    const float* Ap = &As[mrow * 64 + 2 * kh];
    for (int kk = 0; kk < 8; ++kk) {
      float4 bq = wp[(size_t)kk * 32];
      v2f a0 = *(const v2f*)(Ap + 8 * kk);
      v2f a1 = *(const v2f*)(Ap + 8 * kk + 4);
      v2f b0; b0.x = bq.x; b0.y = bq.y;
      v2f b1; b1.x = bq.z; b1.y = bq.w;
      acc = wmma4(a0, b0, acc);
      acc = wmma4(a1, b1, acc);
    }
    for (int r = 0; r < 8; ++r)
      h[(size_t)(r0 + r + 8 * kh) * 256 + col] = acc[r];
  }
}

// h_out[dst] += h[src] * a   (edge softmax weight a = ex/denom[dst])
__global__ __launch_bounds__(256)
void scatter_kernel(const int* __restrict__ src, const int* __restrict__ dst,
                    const float* __restrict__ exb, const float* __restrict__ denom,
                    const float* __restrict__ h, float* __restrict__ h_out) {
  __shared__ float a4[4];
  const int e = blockIdx.x;
  const int tid = threadIdx.x;
  const int d = dst[e];
  if (tid < 4) a4[tid] = exb[(size_t)e * NHd + tid] / denom[(size_t)d * NHd + tid];
  __syncthreads();
  const int s = src[e];
  float v = h[(size_t)s * 256 + tid] * a4[tid >> 6];
  atomicAdd(&h_out[(size_t)d * 256 + tid], v);
}

// new_n = elu((h_out + ebias) @ mnW + mnb) + nf   (K=256, N=64)
__global__ __launch_bounds__(128)
void node_update_kernel(const float* __restrict__ h_out, const float* __restrict__ ebias,
                        const float4* __restrict__ Wp, const float* __restrict__ bn,
                        const float* __restrict__ nf_old, float* __restrict__ nf_new) {
  __shared__ float As[16 * 256];
  const int tid = threadIdx.x;
  const int r0 = blockIdx.x * 16;
  for (int idx = tid; idx < 16 * 256; idx += 128)
    As[idx] = h_out[(size_t)r0 * 256 + idx] + ebias[idx & 255];
  __syncthreads();
  const int lane = tid & 31, wave = tid >> 5, mrow = lane & 15, kh = lane >> 4;
  const int col = wave * 16 + mrow;
  const float4* wp = Wp + ((size_t)wave * 32) * 32 + lane;     // K=256 -> 32 steps
  v8f acc = {};
  const float* Ap = &As[mrow * 256 + 2 * kh];
  for (int kk = 0; kk < 32; ++kk) {
    float4 bq = wp[(size_t)kk * 32];
    v2f a0 = *(const v2f*)(Ap + 8 * kk);
    v2f a1 = *(const v2f*)(Ap + 8 * kk + 4);
    v2f b0; b0.x = bq.x; b0.y = bq.y;
    v2f b1; b1.x = bq.z; b1.y = bq.w;
    acc = wmma4(a0, b0, acc);
    acc = wmma4(a1, b1, acc);
  }
  for (int r = 0; r < 8; ++r) {
    int n = r0 + r + 8 * kh;
    float f = elu1(acc[r] + bn[col]);
    nf_new[(size_t)n * Hd + col] = f + nf_old[(size_t)n * Hd + col];
  }
}

// ------------------------------------------------------------------- readout
__global__ void reduce_feat_kernel(const float* __restrict__ feat, const int* __restrict__ gid,
                                   float* __restrict__ gsum, float* __restrict__ cnt, int nrows) {
  int i = blockIdx.x * blockDim.x + threadIdx.x;
  if (i >= nrows * Hd) return;
  int r = i >> 6, h = i & 63;
  int g = gid[r];
  atomicAdd(&gsum[(size_t)g * Hd + h], feat[i]);
  if (h == 0) atomicAdd(&cnt[g], 1.0f);
}

__global__ void graph_out_kernel(const float* __restrict__ gn, const float* __restrict__ ge,
                                 const float* __restrict__ cn, const float* __restrict__ ce,
                                 const float* __restrict__ fpe, const float* __restrict__ Wg,
                                 const float* __restrict__ bg, float* __restrict__ gout) {
  int i = blockIdx.x * blockDim.x + threadIdx.x;
  if (i >= Bb * Hd) return;
  int b = i >> 6, h = i & 63;
  float invn = 1.0f / fmaxf(cn[b], 1.0f);
  float inve = 1.0f / fmaxf(ce[b], 1.0f);
  float s = bg[h];
  for (int j = 0; j < Hd; ++j) s += gn[b * Hd + j] * invn * Wg[j * Hd + h];
  for (int j = 0; j < Hd; ++j) s += ge[b * Hd + j] * inve * Wg[(Hd + j) * Hd + h];
  for (int j = 0; j < Hd; ++j) s += fpe[b * Hd + j] * Wg[(2 * Hd + j) * Hd + h];
  gout[i] = elu1(s);
}

// ---------------------------------------------------------------------------
extern "C" void kernel_launch(void* const* d_in, const int* in_sizes, int n_in,
                              void* d_out, int out_size, void* d_ws, size_t ws_size,
                              hipStream_t stream) {
  (void)in_sizes; (void)n_in; (void)out_size; (void)ws_size;
  const float* x          = (const float*)d_in[0];
  const float* e          = (const float*)d_in[1];
  const int*   placed_rc  = (const int*)d_in[2];
  const float* fp         = (const float*)d_in[3];
  const int*   src        = (const int*)d_in[4];
  const int*   dst        = (const int*)d_in[5];
  const int*   node_gid   = (const int*)d_in[6];
  const int*   edge_gid   = (const int*)d_in[7];
  const float* W_fp       = (const float*)d_in[8];
  const float* b_fp       = (const float*)d_in[9];
  const float* W_n0       = (const float*)d_in[10];
  const float* b_n0       = (const float*)d_in[11];
  const float* W_e0       = (const float*)d_in[12];
  const float* b_e0       = (const float*)d_in[13];
  const float* emb_rc     = (const float*)d_in[14];
  const float* fc_node_W  = (const float*)d_in[15];
  const float* fc_ni_W    = (const float*)d_in[16];
  const float* fc_nj_W    = (const float*)d_in[17];
  const float* fc_fij_W   = (const float*)d_in[18];
  const float* attn       = (const float*)d_in[19];
  const float* egat_bias  = (const float*)d_in[20];
  const float* mlp_n_W    = (const float*)d_in[21];
  const float* mlp_n_b    = (const float*)d_in[22];
  const float* mlp_e_W    = (const float*)d_in[23];
  const float* mlp_e_b    = (const float*)d_in[24];
  const float* W_g        = (const float*)d_in[25];
  const float* b_g        = (const float*)d_in[26];

  float* out = (float*)d_out;          // [N*H nf | B*H g]

  // Workspace layout (f32 words)
  float* ws = (float*)d_ws;
  size_t off = 0;
  float* nfA = ws + off;  off += (size_t)Nn * Hd;
  float* nfB = ws + off;  off += (size_t)Nn * Hd;
  float* efA = ws + off;  off += (size_t)Ee * Hd;
  float* efB = ws + off;  off += (size_t)Ee * Hd;
  float* exb = ws + off;  off += (size_t)Ee * NHd;     // logits -> ex (in place)
  unsigned* m_u = (unsigned*)(ws + off); off += (size_t)Nn * NHd;
  float* denom  = ws + off; off += (size_t)Nn * NHd;
  float* h_out  = ws + off; off += (size_t)Nn * 256;   // m_u..h_out contiguous (one zero pass)
  float* hbuf   = ws + off; off += (size_t)Nn * 256;
  float* fpe    = ws + off; off += Bb * Hd;
  float* gn     = ws + off; off += Bb * Hd;            // gn..ce contiguous (one zero pass)
  float* ge     = ws + off; off += Bb * Hd;
  float* cn     = ws + off; off += Bb;
  float* ce     = ws + off; off += Bb;
  // Packed weight fragments (float, 16B-aligned offsets)
  float* WcatP  = ws + off; off += (size_t)Ld * 192 * 256;  // 49152/layer
  float* WmeP   = ws + off; off += (size_t)Ld * 256 * 64;
  float* WnodeP = ws + off; off += (size_t)Ld * 64 * 256;
  float* WmnP   = ws + off; off += (size_t)Ld * 256 * 64;

  // One-time weight packing into WMMA B-fragment order
  for (int l = 0; l < Ld; ++l) {
    pack_wcat_kernel<<<(12288 + 255) / 256, 256, 0, stream>>>(
        fc_ni_W + (size_t)l * 64 * 256, fc_nj_W + (size_t)l * 64 * 256,
        fc_fij_W + (size_t)l * 64 * 256, WcatP + (size_t)l * 192 * 256);
    pack_w_kernel<<<(4096 + 255) / 256, 256, 0, stream>>>(
        mlp_e_W + (size_t)l * 256 * 64, WmeP + (size_t)l * 256 * 64, 256, 64);
    pack_w_kernel<<<(4096 + 255) / 256, 256, 0, stream>>>(
        fc_node_W + (size_t)l * 64 * 256, WnodeP + (size_t)l * 64 * 256, 64, 256);
    pack_w_kernel<<<(4096 + 255) / 256, 256, 0, stream>>>(
        mlp_n_W + (size_t)l * 256 * 64, WmnP + (size_t)l * 256 * 64, 256, 64);
  }

  // Input encoders
  fp_enc_kernel<<<(Bb * Hd + 255) / 256, 256, 0, stream>>>(fp, W_fp, b_fp, fpe);
  node_init_kernel<<<(Nn * Hd + 255) / 256, 256, 0, stream>>>(x, placed_rc, W_n0, b_n0, emb_rc, nfA);
  edge_init_kernel<<<(Ee * Hd + 255) / 256, 256, 0, stream>>>(e, W_e0, b_e0, efA);

  for (int l = 0; l < Ld; ++l) {
    const float* nf_in  = (l == 0) ? nfA : nfB;
    float*       nf_out = (l == 0) ? nfB : out;        // final nf goes straight to d_out
    const float* ef_in  = (l == 0) ? efA : efB;
    float*       ef_out = (l == 0) ? efB : efA;        // final ef lands in efA

    // zero m_u + denom + h_out (contiguous 32-bit words)
    zero_u32_kernel<<<2048, 256, 0, stream>>>(m_u, Nn * NHd * 2 + Nn * 256);

    egat_edge_kernel<<<Ee / 16, 128, 0, stream>>>(
        nf_in, ef_in, src, dst,
        (const float4*)(WcatP + (size_t)l * 192 * 256),
        attn + (size_t)l * NHd * Hd,
        (const float4*)(WmeP + (size_t)l * 256 * 64),
        mlp_e_b + (size_t)l * Hd,
        ef_out, exb, m_u);

    softmax_exp_kernel<<<(Ee * NHd + 255) / 256, 256, 0, stream>>>(dst, m_u, exb, denom);

    node_h_kernel<<<Nn / 16, 128, 0, stream>>>(
        nf_in, (const float4*)(WnodeP + (size_t)l * 64 * 256), hbuf);

    scatter_kernel<<<Ee, 256, 0, stream>>>(src, dst, exb, denom, hbuf, h_out);

    node_update_kernel<<<Nn / 16, 128, 0, stream>>>(
        h_out,
        egat_bias + (size_t)l * 256,
        (const float4*)(WmnP + (size_t)l * 256 * 64),
        mlp_n_b + (size_t)l * Hd,
        nf_in, nf_out);
  }

  // Readout: zero accumulators (gn|ge|cn|ce contiguous), segment means, final MLP
  zero_u32_kernel<<<64, 256, 0, stream>>>((unsigned*)gn, Bb * Hd * 2 + Bb * 2);
  reduce_feat_kernel<<<(Nn * Hd + 255) / 256, 256, 0, stream>>>(out, node_gid, gn, cn, Nn);
  reduce_feat_kernel<<<(Ee * Hd + 255) / 256, 256, 0, stream>>>(efA, edge_gid, ge, ce, Ee);
  graph_out_kernel<<<(Bb * Hd + 255) / 256, 256, 0, stream>>>(gn, ge, cn, ce, fpe, W_g, b_g,
                                                              out + (size_t)Nn * Hd);
}